// ComputeGsct_28123445854649
// MI455X (gfx1250) — compile-verified
//
#include <hip/hip_runtime.h>
#include <hip/hip_bf16.h>

// ---------------------------------------------------------------------------
// Gsct = sum_n (Gsr * kai) @ Grf  (complex 3x3 per voxel, reduced over N)
// Memory-bound streaming reduction (~637 MB @ 23.3 TB/s => ~27 us floor).
// Kernel 1: grid-strided per-lane accumulation (18 f32 acc), shfl+LDS block
//           reduction, deterministic partials to d_ws.
// Kernel 2: ones-matrix GEMM reduction of partials via V_WMMA_F32_16X16X4_F32.
// ---------------------------------------------------------------------------

typedef __attribute__((ext_vector_type(2))) float v2f;
typedef __attribute__((ext_vector_type(8))) float v8f;

#define BLOCK_T 256
#define MAX_G   64

// Physics constants (double at compile time, cast to f32 like the reference).
constexpr double kPI    = 3.141592653589793;
constexpr double kC     = 299792458.0;
constexpr double kFREQ  = 2.4e9;
constexpr double kOMEGA = 2.0 * kPI * kFREQ;
constexpr double kMU0   = 4.0 * kPI * 1e-7;
constexpr double kEPS0  = 1.0 / (kC * kC * kMU0);
constexpr float  KAI_R_SCALE = 10.0f;
constexpr float  KAI_I_SCALE = (float)(-1.0 / (kOMEGA * kEPS0));

// ---------------------------------------------------------------------------
// Kernel 1: streaming accumulation. grid = (gcount, B). Each block produces one
// 18-float partial: part[(b*gcount + g)*18 + e], e = comp*2 + {re,im}.
// ---------------------------------------------------------------------------
__global__ void __launch_bounds__(BLOCK_T)
gsct_partial_kernel(const float* __restrict__ x0,   // (B,N,9,2) Gsr
                    const float* __restrict__ x1,   // (B,N,9,2) Grf
                    const float* __restrict__ x2,   // (B,N,2)   kai
                    float* __restrict__ part,
                    int nvox, int gcount) {
  const int b   = blockIdx.y;
  const int g   = blockIdx.x;
  const int tid = threadIdx.x;

  float cr[9], ci[9];
#pragma unroll
  for (int e = 0; e < 9; ++e) { cr[e] = 0.0f; ci[e] = 0.0f; }

  const v2f* __restrict__ gsr = (const v2f*)x0;  // (re,im) pairs
  const v2f* __restrict__ grf = (const v2f*)x1;
  const v2f* __restrict__ kai = (const v2f*)x2;

  const long long base   = (long long)b * (long long)nvox;
  const int       stride = gcount * BLOCK_T;

  for (int n = g * BLOCK_T + tid; n < nvox; n += stride) {
    const long long v = base + n;

    const v2f   k2 = __builtin_nontemporal_load(&kai[v]);
    const float kr = k2.x * KAI_R_SCALE;
    const float ki = k2.y * KAI_I_SCALE;

    v2f gv[9], hv[9];
    const v2f* __restrict__ gp = &gsr[v * 9];
    const v2f* __restrict__ hp = &grf[v * 9];
#pragma unroll
    for (int e = 0; e < 9; ++e) gv[e] = __builtin_nontemporal_load(&gp[e]);
#pragma unroll
    for (int e = 0; e < 9; ++e) hv[e] = __builtin_nontemporal_load(&hp[e]);

    // A = Gsr * kai (elementwise complex scale)
    float ar[9], ai[9];
#pragma unroll
    for (int e = 0; e < 9; ++e) {
      ar[e] = gv[e].x * kr - gv[e].y * ki;
      ai[e] = gv[e].y * kr + gv[e].x * ki;
    }

    // C += A @ Grf (complex 3x3)
#pragma unroll
    for (int i = 0; i < 3; ++i) {
#pragma unroll
      for (int k = 0; k < 3; ++k) {
        float sr = cr[i * 3 + k];
        float si = ci[i * 3 + k];
#pragma unroll
        for (int j = 0; j < 3; ++j) {
          const float arr = ar[i * 3 + j];
          const float aii = ai[i * 3 + j];
          const v2f   h   = hv[j * 3 + k];
          sr = __builtin_fmaf(arr, h.x, sr);
          sr = __builtin_fmaf(-aii, h.y, sr);
          si = __builtin_fmaf(aii, h.x, si);
          si = __builtin_fmaf(arr, h.y, si);
        }
        cr[i * 3 + k] = sr;
        ci[i * 3 + k] = si;
      }
    }
  }

  // Wave32 butterfly reduction (deterministic).
#pragma unroll
  for (int e = 0; e < 9; ++e) {
    for (int off = 16; off >= 1; off >>= 1) {
      cr[e] += __shfl_xor(cr[e], off, 32);
      ci[e] += __shfl_xor(ci[e], off, 32);
    }
  }

  // Cross-wave reduction through LDS (8 waves per block).
  __shared__ float lds[BLOCK_T / 32][18];
  const int lane = tid & 31;
  const int w    = tid >> 5;
  if (lane == 0) {
#pragma unroll
    for (int e = 0; e < 9; ++e) {
      lds[w][2 * e + 0] = cr[e];
      lds[w][2 * e + 1] = ci[e];
    }
  }
  __syncthreads();

  if (tid < 18) {
    float s = 0.0f;
#pragma unroll
    for (int w2 = 0; w2 < BLOCK_T / 32; ++w2) s += lds[w2][tid];
    part[((long long)(b * gcount + g)) * 18 + tid] = s;
  }
}

// ---------------------------------------------------------------------------
// Kernel 2: per batch, reduce gcount partial vectors of 18 floats with chained
// V_WMMA_F32_16X16X4_F32: C += A(16x4) * ones(4x16).
//   A[e][k] = P[4t + k][e]   (e = element 0..15, k = partial-slot within step)
// A operand layout (ISA 7.12.2): lanes 0-15 hold K=0 (VGPR0), K=1 (VGPR1);
// lanes 16-31 hold K=2, K=3  -> lane L gathers e = L&15, gsel = (L>>4)*2.
// C layout: VGPR r = row M=r (lanes 0-15) / M=8+r (lanes 16-31), all columns
// identical since B is all-ones -> lane 0 emits e=0..7, lane 16 emits e=8..15.
// One wave per batch; EXEC is all-ones through the WMMA chain.
// ---------------------------------------------------------------------------
__global__ void __launch_bounds__(32)
gsct_final_kernel(const float* __restrict__ part, float* __restrict__ out,
                  int gcount) {
  const int b    = blockIdx.x;
  const int lane = threadIdx.x;            // 0..31, all active
  const int e    = lane & 15;
  const int gsel = (lane >> 4) << 1;       // 0 or 2

  const float* __restrict__ pb = part + (long long)b * gcount * 18;

  v8f c = {};                              // zero accumulator
  const v2f ones = {1.0f, 1.0f};           // B = ones(4x16)

  const int steps = gcount >> 2;           // gcount is a multiple of 4
  for (int t = 0; t < steps; ++t) {
    v2f a;
    a.x = pb[(4 * t + gsel + 0) * 18 + e];
    a.y = pb[(4 * t + gsel + 1) * 18 + e];
    // D = A * ones + C  : row-sums of A accumulate the partials per element.
    c = __builtin_amdgcn_wmma_f32_16x16x4_f32(
        /*neg_a=*/false, a, /*neg_b=*/false, ones,
        /*c_mod=*/(short)0, c, /*reuse_a=*/false, /*reuse_b=*/false);
  }

  float* __restrict__ ob = out + b * 18;
  if (lane == 0) {
#pragma unroll
    for (int r = 0; r < 8; ++r) ob[r] = c[r];        // elements 0..7
    float s16 = 0.0f, s17 = 0.0f;                    // tail: elements 16,17
    for (int g2 = 0; g2 < gcount; ++g2) {
      s16 += pb[g2 * 18 + 16];
      s17 += pb[g2 * 18 + 17];
    }
    ob[16] = s16;
    ob[17] = s17;
  } else if (lane == 16) {
#pragma unroll
    for (int r = 0; r < 8; ++r) ob[8 + r] = c[r];    // elements 8..15
  }
}

// ---------------------------------------------------------------------------
extern "C" void kernel_launch(void* const* d_in, const int* in_sizes, int n_in,
                              void* d_out, int out_size, void* d_ws, size_t ws_size,
                              hipStream_t stream) {
  const float* x0 = (const float*)d_in[0];
  const float* x1 = (const float*)d_in[1];
  const float* x2 = (const float*)d_in[2];
  float* out  = (float*)d_out;
  float* part = (float*)d_ws;

  const int Bn   = out_size / 18;                 // 32
  const int nvox = in_sizes[2] / (Bn * 2);        // 131072

  // Partials: Bn * gcount * 18 floats in d_ws; clamp gcount to fit (mult of 4).
  int gcount = MAX_G;
  const size_t per_g = (size_t)Bn * 18 * sizeof(float);
  if ((size_t)gcount * per_g > ws_size) {
    gcount = (int)(ws_size / per_g);
    gcount &= ~3;
    if (gcount < 4) gcount = 4;
  }

  dim3 grid1(gcount, Bn);
  gsct_partial_kernel<<<grid1, BLOCK_T, 0, stream>>>(x0, x1, x2, part, nvox, gcount);
  gsct_final_kernel<<<Bn, 32, 0, stream>>>(part, out, gcount);
}